// GNN_6184752906609
// MI455X (gfx1250) — compile-verified
//
#include <hip/hip_runtime.h>

typedef float v2f __attribute__((ext_vector_type(2)));
typedef float v8f __attribute__((ext_vector_type(8)));

#define N_NODES 100000
#define NGRAPH  128
#define MTILES  (N_NODES / 16)   // 6250 row tiles (exact)
#define WPB     2                // waves per block
#define GEMM_GX (MTILES / WPB)   // 3125

// ---------------------------------------------------------------------------
// Fused tile GEMM:  out[16x16 tile] = relu?( A@W (+ A2@W2) (+ addend) (+ bias) )
// One wave32 per 16x16 tile, f32 WMMA 16x16x4, K fully unrolled.
// A fragment (16x4 f32): lane<16 -> M=lane, VGPR0=K0,VGPR1=K1 ; lane>=16 -> K2,K3
// B fragment (4x16 f32): mirrored (row K striped across lanes)
// C/D: VGPR r -> row r+8*half, col lane&15
// ---------------------------------------------------------------------------
template<int K1, int K2, int DO, bool RELU, bool ADDEND, bool BIAS>
__global__ __launch_bounds__(32 * WPB) void gnn_gemm_kernel(
    const float* __restrict__ A,   const float* __restrict__ W,
    const float* __restrict__ A2,  const float* __restrict__ W2,
    const float* __restrict__ add, const float* __restrict__ bias,
    float* __restrict__ out)
{
    const int lane = threadIdx.x & 31;
    const int wave = threadIdx.x >> 5;
    const int half = lane >> 4;       // 0: lanes 0-15, 1: lanes 16-31
    const int mn   = lane & 15;
    const int m0   = (blockIdx.x * WPB + wave) * 16;
    const int n0   = blockIdx.y * 16;

    v8f c;
#pragma unroll
    for (int r = 0; r < 8; ++r) {
        float v = BIAS ? bias[n0 + mn] : 0.0f;
        if (ADDEND) v += add[(m0 + r + 8 * half) * DO + n0 + mn];
        c[r] = v;
    }

    {
        const float* arow = A + (m0 + mn) * K1;
#pragma unroll
        for (int k = 0; k < K1; k += 4) {
            v2f a, b;
            a[0] = arow[k + 2 * half + 0];
            a[1] = arow[k + 2 * half + 1];
            b[0] = W[(k + 2 * half + 0) * DO + n0 + mn];
            b[1] = W[(k + 2 * half + 1) * DO + n0 + mn];
            c = __builtin_amdgcn_wmma_f32_16x16x4_f32(false, a, false, b,
                                                      (short)0, c, false, false);
        }
    }
    if constexpr (K2 > 0) {
        const float* arow = A2 + (m0 + mn) * K2;
#pragma unroll
        for (int k = 0; k < K2; k += 4) {
            v2f a, b;
            a[0] = arow[k + 2 * half + 0];
            a[1] = arow[k + 2 * half + 1];
            b[0] = W2[(k + 2 * half + 0) * DO + n0 + mn];
            b[1] = W2[(k + 2 * half + 1) * DO + n0 + mn];
            c = __builtin_amdgcn_wmma_f32_16x16x4_f32(false, a, false, b,
                                                      (short)0, c, false, false);
        }
    }

#pragma unroll
    for (int r = 0; r < 8; ++r) {
        float v = c[r];
        if (RELU) v = fmaxf(v, 0.0f);
        out[(m0 + r + 8 * half) * DO + n0 + mn] = v;
    }
}

// ---------------------------------------------------------------------------
// Edge scatter-add: agg[dst[e], f] += feat[src[e], f]   (F = 1<<LOGF floats)
// Consecutive threads share an edge -> coalesced feature reads, s/d broadcast.
// Non-returning f32 atomics -> global_atomic_add_f32 (STOREcnt, fire&forget).
// ---------------------------------------------------------------------------
template<int LOGF>
__global__ void gnn_scatter_kernel(const float* __restrict__ feat,
                                   const int* __restrict__ srcv,
                                   const int* __restrict__ dstv,
                                   float* __restrict__ agg, int nedges)
{
    const int F = 1 << LOGF;
    unsigned tid   = blockIdx.x * blockDim.x + threadIdx.x;
    unsigned total = (unsigned)nedges << LOGF;
    if (tid >= total) return;
    int e = (int)(tid >> LOGF);
    int f = (int)(tid & (F - 1));
    int s = srcv[e];
    int d = dstv[e];
    atomicAdd(&agg[d * F + f], feat[s * F + f]);
}

__global__ void gnn_fill_zero(float* __restrict__ p, int n)
{
    int i = blockIdx.x * blockDim.x + threadIdx.x;
    if (i < n) p[i] = 0.0f;
}

// sums[batch[i], f] += h[i, f]; cnts[batch[i]] += 1  (64 feats)
__global__ void gnn_pool_kernel(const float* __restrict__ h,
                                const int* __restrict__ batch,
                                float* __restrict__ sums,
                                float* __restrict__ cnts)
{
    int tid = blockIdx.x * blockDim.x + threadIdx.x;
    if (tid >= N_NODES * 64) return;
    int i = tid >> 6;
    int f = tid & 63;
    int b = batch[i];
    atomicAdd(&sums[(b << 6) + f], h[tid]);
    if (f == 0) atomicAdd(&cnts[b], 1.0f);
}

__global__ void gnn_finalize_kernel(const float* __restrict__ sums,
                                    const float* __restrict__ cnts,
                                    const float* __restrict__ W2,
                                    const float* __restrict__ b2,
                                    float* __restrict__ out)
{
    int g = threadIdx.x;
    if (g >= NGRAPH) return;
    float inv = 1.0f / fmaxf(cnts[g], 1.0f);
    float acc = b2[0];
#pragma unroll
    for (int f = 0; f < 64; ++f)
        acc = fmaf(sums[(g << 6) + f] * inv, W2[f], acc);
    out[g] = acc;
}

extern "C" void kernel_launch(void* const* d_in, const int* in_sizes, int n_in,
                              void* d_out, int out_size, void* d_ws, size_t ws_size,
                              hipStream_t stream)
{
    const float* x      = (const float*)d_in[0];
    const int*   eidx   = (const int*)d_in[1];
    const int*   batch  = (const int*)d_in[2];
    const float* Wrel1  = (const float*)d_in[4];
    const float* brel1  = (const float*)d_in[5];
    const float* Wroot1 = (const float*)d_in[6];
    const float* Wrel2  = (const float*)d_in[7];
    const float* brel2  = (const float*)d_in[8];
    const float* Wroot2 = (const float*)d_in[9];
    const float* Wrel3  = (const float*)d_in[10];
    const float* brel3  = (const float*)d_in[11];
    const float* Wroot3 = (const float*)d_in[12];
    const float* Wrel4  = (const float*)d_in[13];
    const float* brel4  = (const float*)d_in[14];
    const float* Wroot4 = (const float*)d_in[15];
    const float* W1     = (const float*)d_in[16];
    const float* b1     = (const float*)d_in[17];
    const float* W2     = (const float*)d_in[18];
    const float* b2     = (const float*)d_in[19];

    const int E = in_sizes[1] / 2;
    const int* srcv = eidx;          // edge_index row 0
    const int* dstv = eidx + E;      // edge_index row 1

    // Workspace layout (floats): agg[N*64] | hA[N*128] | hB[N*128] | sums[G*64] | cnts[G]
    float* agg  = (float*)d_ws;
    float* hA   = agg  + N_NODES * 64;
    float* hB   = hA   + N_NODES * 128;
    float* sums = hB   + N_NODES * 128;
    float* cnts = sums + NGRAPH * 64;

    const dim3 blk(32 * WPB);
    auto cdiv = [](int a, int b) { return (a + b - 1) / b; };

    // ---- Layer 1 (64 -> 16): transform-first (scatter in 16-dim, 4x less traffic)
    // y1 = x @ Wrel1  -> hB
    gnn_gemm_kernel<64, 0, 16, false, false, false>
        <<<dim3(GEMM_GX, 1), blk, 0, stream>>>(x, Wrel1, nullptr, nullptr, nullptr, nullptr, hB);
    gnn_fill_zero<<<cdiv(N_NODES * 16, 256), 256, 0, stream>>>(agg, N_NODES * 16);
    gnn_scatter_kernel<4><<<cdiv(E * 16, 256), 256, 0, stream>>>(hB, srcv, dstv, agg, E);
    // h1 = relu(agg + x @ Wroot1 + brel1) -> hA   (agg rides in the WMMA accumulator)
    gnn_gemm_kernel<64, 0, 16, true, true, true>
        <<<dim3(GEMM_GX, 1), blk, 0, stream>>>(x, Wroot1, nullptr, nullptr, agg, brel1, hA);

    // ---- Layer 2 (16 -> 32): aggregate-first
    gnn_fill_zero<<<cdiv(N_NODES * 16, 256), 256, 0, stream>>>(agg, N_NODES * 16);
    gnn_scatter_kernel<4><<<cdiv(E * 16, 256), 256, 0, stream>>>(hA, srcv, dstv, agg, E);
    gnn_gemm_kernel<16, 16, 32, true, false, true>
        <<<dim3(GEMM_GX, 2), blk, 0, stream>>>(agg, Wrel2, hA, Wroot2, nullptr, brel2, hB);

    // ---- Layer 3 (32 -> 64)
    gnn_fill_zero<<<cdiv(N_NODES * 32, 256), 256, 0, stream>>>(agg, N_NODES * 32);
    gnn_scatter_kernel<5><<<cdiv(E * 32, 256), 256, 0, stream>>>(hB, srcv, dstv, agg, E);
    gnn_gemm_kernel<32, 32, 64, true, false, true>
        <<<dim3(GEMM_GX, 4), blk, 0, stream>>>(agg, Wrel3, hB, Wroot3, nullptr, brel3, hA);

    // ---- Layer 4 (64 -> 128)
    gnn_fill_zero<<<cdiv(N_NODES * 64, 256), 256, 0, stream>>>(agg, N_NODES * 64);
    gnn_scatter_kernel<6><<<cdiv(E * 64, 256), 256, 0, stream>>>(hA, srcv, dstv, agg, E);
    gnn_gemm_kernel<64, 64, 128, true, false, true>
        <<<dim3(GEMM_GX, 8), blk, 0, stream>>>(agg, Wrel4, hA, Wroot4, nullptr, brel4, hB);

    // ---- Head: h5 = h4 @ W1 + b1  (128 -> 64) -> hA
    gnn_gemm_kernel<128, 0, 64, false, false, true>
        <<<dim3(GEMM_GX, 4), blk, 0, stream>>>(hB, W1, nullptr, nullptr, nullptr, b1, hA);

    // ---- Mean pool per graph + final projection
    gnn_fill_zero<<<cdiv(NGRAPH * 64 + NGRAPH, 256), 256, 0, stream>>>(sums, NGRAPH * 64 + NGRAPH);
    gnn_pool_kernel<<<cdiv(N_NODES * 64, 256), 256, 0, stream>>>(hA, batch, sums, cnts);
    gnn_finalize_kernel<<<1, NGRAPH, 0, stream>>>(sums, cnts, W2, b2, (float*)d_out);
}